// SecAwareBranch_87883620811000
// MI455X (gfx1250) — compile-verified
//
#include <hip/hip_runtime.h>

typedef __attribute__((ext_vector_type(2))) float v2f;
typedef __attribute__((ext_vector_type(4))) float v4f;
typedef __attribute__((ext_vector_type(8))) float v8f;

#define NBATCH    4096
#define NF        64
#define ND        128
#define ROWW      132                 // padded LDS row stride in floats (528 B)
#define TILE_FL   (NF * ROWW)         // 8448 floats per LDS tile
#define THREADS   256
#define NB_PER_BLK 4
#define NBLOCKS   (NBATCH / NB_PER_BLK)

__device__ __forceinline__ unsigned lds_addr_of(const void* p) {
  // flat shared pointer: low 32 bits are the group-segment-relative offset
  return (unsigned)(unsigned long long)p;
}

// Stage one 64x128 f32 batch tile (32 KB) into LDS via async B128 copies.
// Each thread issues 8 per-lane 16B transfers; GVS mode (saddr = batch base).
__device__ __forceinline__ void issue_batch_async(const float* xb, float* dst, int t) {
#pragma unroll
  for (int i = 0; i < 8; ++i) {
    const int g    = t + THREADS * i;     // float4 index within batch [0,2048)
    const int row  = g >> 5;
    const int col4 = g & 31;
    unsigned lds  = lds_addr_of(&dst[row * ROWW + col4 * 4]);
    unsigned goff = (unsigned)(g * 16);   // byte offset, max 32752
    asm volatile("global_load_async_to_lds_b128 %0, %1, %2"
                 :: "v"(lds), "v"(goff), "s"(xb)
                 : "memory");
  }
}

__global__ __launch_bounds__(THREADS) void fm_cross_kernel(
    const float* __restrict__ x, float* __restrict__ out) {
  __shared__ __align__(16) float sx[2][TILE_FL];
  __shared__ __align__(16) float ssum[ND];

  const int t    = (int)threadIdx.x;
  const int lane = t & 31;
  const int wave = t >> 5;        // 8 waves -> 8 column tiles of 16
  const int c    = lane & 15;     // column within tile (N = lane % 16)
  const int h    = lane >> 4;     // lane half selects k rows

  const int b0 = (int)blockIdx.x * NB_PER_BLK;
  issue_batch_async(x + (size_t)b0 * (NF * ND), sx[0], t);

  int cur = 0;
#pragma unroll 1
  for (int it = 0; it < NB_PER_BLK; ++it) {
    const int b = b0 + it;

    asm volatile("s_wait_asynccnt 0" ::: "memory");
    __syncthreads();   // buf[cur] fully written; all waves done with prev compute

    if (it + 1 < NB_PER_BLK)
      issue_batch_async(x + (size_t)(b + 1) * (NF * ND), sx[cur ^ 1], t);

    // ---- field_sum via V_WMMA_F32_16X16X4_F32: A = ones(16x4) => column sums.
    // Wave handles d-columns [wave*16, wave*16+16); 16 WMMA steps cover 64 rows.
    // With A all-ones the k-slot assignment inside B is sum-invariant; only
    // n = lane%16 matters (matches B/C/D row-striped-across-lanes layout).
    const float* base = &sx[cur][wave * 16 + c];
    v8f acc = {};
    v2f ones = {1.0f, 1.0f};
#pragma unroll
    for (int k = 0; k < 16; ++k) {
      const int r0 = 4 * k + 2 * h;        // rows {4k,4k+1,4k+2,4k+3} over h & slots
      v2f bmat;
      bmat[0] = base[r0 * ROWW];
      bmat[1] = base[(r0 + 1) * ROWW];
      acc = __builtin_amdgcn_wmma_f32_16x16x4_f32(
          /*neg_a=*/false, ones, /*neg_b=*/false, bmat,
          /*c_mod=*/(short)0, acc, /*reuse_a=*/false, /*reuse_b=*/false);
    }
    // Every row of D equals the column sum; lane<16 holds (M=0, N=lane) in acc[0].
    if (lane < 16) ssum[wave * 16 + lane] = acc[0];
    __syncthreads();   // sums visible to all waves

    // ---- elementwise expand: out = x*s - x*x, coalesced B128 NT stores.
    float* outb = out + (size_t)b * (NF * ND);
#pragma unroll
    for (int i = 0; i < 8; ++i) {
      const int g    = t + THREADS * i;
      const int row  = g >> 5;
      const int col4 = g & 31;
      const v4f xv = *reinterpret_cast<const v4f*>(&sx[cur][row * ROWW + col4 * 4]);
      const v4f sv = *reinterpret_cast<const v4f*>(&ssum[col4 * 4]);
      v4f o = xv * sv - xv * xv;
      __builtin_nontemporal_store(o, reinterpret_cast<v4f*>(outb) + g);
    }
    cur ^= 1;
  }
}

extern "C" void kernel_launch(void* const* d_in, const int* in_sizes, int n_in,
                              void* d_out, int out_size, void* d_ws, size_t ws_size,
                              hipStream_t stream) {
  const float* x = (const float*)d_in[0];
  float* out = (float*)d_out;
  (void)in_sizes; (void)n_in; (void)out_size; (void)d_ws; (void)ws_size;
  fm_cross_kernel<<<dim3(NBLOCKS), dim3(THREADS), 0, stream>>>(x, out);
}